// CrossDomainAnalogy_34359739034
// MI455X (gfx1250) — compile-verified
//
#include <hip/hip_runtime.h>
#include <math.h>

// ---------------------------------------------------------------------------
// CrossDomainAnalogy for MI455X (gfx1250, wave32).
// Algebraic folding removes the huge condenser GEMMs; pair-stage batched
// GEMMs (M=4, padded to 16) run on V_WMMA_F32_16X16X4_F32 in fp32.
// ---------------------------------------------------------------------------

typedef __attribute__((ext_vector_type(2))) float v2f;
typedef __attribute__((ext_vector_type(8))) float v8f;

#define Bz 4
#define Tz 2048
#define Dz 1024
#define Ez 6
#define Pz 36
#define DAz 512
#define THRESH 0.3f
#define LN_EPS 1e-5f
#define SCALE 0.03125f   /* 1024^-0.5 */

// ---- workspace layout (floats) --------------------------------------------
#define OFF_QK      ((size_t)0)                         // 6*1024
#define OFF_QB      (OFF_QK + 6144)                     // 6 (+pad)
#define OFF_AW      (OFF_QB + 64)                       // E*B*T = 49152 (logits, then attn*w in place)
#define OFF_XAP     (OFF_AW + 49152)                    // 24*8*1024
#define OFF_XA      (OFF_XAP + 196608)                  // 24*1024
#define OFF_FORM    (OFF_XA + 24576)                    // 6*4*1024
#define OFF_CMAT    (OFF_FORM + 24576)                  // 36*4*1024
#define OFF_H       (OFF_CMAT + 147456)                 // 36*4*512
#define OFF_HS      (OFF_H + 73728)                     // 36*4*1024
#define OFF_INS     (OFF_HS + 147456)                   // 36*4*1024
#define OFF_STR     (OFF_INS + 147456)                  // 36*4
#define OFF_MASK    (OFF_STR + 160)                     // 36 (+pad)
#define OFF_ACT     (OFF_MASK + 64)                     // 1 (+pad)
#define OFF_NRM     (OFF_ACT + 32)                      // 4*1024
#define OFF_PPART   (OFF_NRM + 4096)                    // 16*1024
#define OFF_ADD     (OFF_PPART + 16384)                 // 4*1024  (16B aligned)

#define OUT_AVG  8388608
#define OUT_ACT  8388644

__device__ __forceinline__ float wave_sum(float v) {
#pragma unroll
  for (int m = 16; m > 0; m >>= 1) v += __shfl_xor(v, m, 32);
  return v;
}

__device__ __forceinline__ float gelu_exact(float x) {
  return 0.5f * x * (1.0f + erff(x * 0.70710678118654752440f));
}

// ---- WMMA main loop: OUT(16xN tile) = Xpad(16xK) * W(KxN), rows 0..3 valid -
__device__ __forceinline__ v8f wmma_rowblock(const float* __restrict__ W, int ldw,
                                             const float* __restrict__ xs,
                                             int K, int n0) {
  const int lane = threadIdx.x & 31;
  const int nidx = lane & 15;          // M row for A, N col for B/D
  const int hi   = lane >> 4;
  const int mrow = nidx;
  const float mval = (mrow < Bz) ? 1.0f : 0.0f;
  v8f acc = {};
  for (int kk = 0; kk < K; kk += 4) {
    const int k0 = kk + 2 * hi;
    v2f a, b;
    a.x = mval * xs[mrow * K + k0];
    a.y = mval * xs[mrow * K + k0 + 1];
    b.x = W[(size_t)k0 * ldw + n0 + nidx];
    b.y = W[(size_t)(k0 + 1) * ldw + n0 + nidx];
    acc = __builtin_amdgcn_wmma_f32_16x16x4_f32(false, a, false, b,
                                                (short)0, acc, false, false);
  }
  return acc;
}

// ===========================================================================
// K0: qbias[e] = q[e] . bk[e]
__global__ void k_qbias(const float* __restrict__ q, const float* __restrict__ bk,
                        float* __restrict__ ws) {
  const int e = threadIdx.x >> 5, lane = threadIdx.x & 31;
  if (e >= Ez) return;
  float acc = 0.f;
  for (int i = 0; i < Dz / 32; ++i) {
    const int f = lane + 32 * i;
    acc += q[e * Dz + f] * bk[e * Dz + f];
  }
  acc = wave_sum(acc);
  if (lane == 0) ws[OFF_QB + e] = acc;
}

// K1: qk[e][d] = sum_f Wk[e][d][f] * q[e][f]
__global__ void k_fold_q(const float* __restrict__ Wk, const float* __restrict__ q,
                         float* __restrict__ ws) {
  const int row = blockIdx.x * 8 + (threadIdx.x >> 5);  // (e,d)
  const int lane = threadIdx.x & 31;
  const int e = row >> 10, d = row & 1023;
  const float* wrow = Wk + ((size_t)e << 20) + ((size_t)d << 10);
  const float* qrow = q + e * Dz;
  float acc = 0.f;
  for (int i = 0; i < Dz / 32; ++i) {
    const int f = lane + 32 * i;
    acc += wrow[f] * qrow[f];
  }
  acc = wave_sum(acc);
  if (lane == 0) ws[OFF_QK + row] = acc;
}

// K2: logits[e,b,t] = SCALE*(w[b,t,e]*(eo[e,b,t].qk[e]) + qbias[e])
__global__ void k_logits(const float* __restrict__ eo, const float* __restrict__ ew,
                         float* __restrict__ ws) {
  const int wid = blockIdx.x * 8 + (threadIdx.x >> 5);  // e*8192 + b*2048 + t
  const int lane = threadIdx.x & 31;
  const int e = wid >> 13, bt = wid & 8191;
  const int b = bt >> 11, t = bt & 2047;
  const float4* x4 = (const float4*)(eo + (size_t)wid * Dz);
  const float4* q4 = (const float4*)(ws + OFF_QK + e * Dz);
  float acc = 0.f;
#pragma unroll
  for (int i = 0; i < 8; ++i) {
    const float4 xv = x4[lane + 32 * i];
    const float4 qv = q4[lane + 32 * i];
    acc += xv.x * qv.x + xv.y * qv.y + xv.z * qv.z + xv.w * qv.w;
  }
  acc = wave_sum(acc);
  if (lane == 0) {
    const float w = ew[(size_t)(b * Tz + t) * Ez + e];
    ws[OFF_AW + wid] = SCALE * (w * acc + ws[OFF_QB + e]);
  }
}

// K3: softmax over T per (e,b); write attn*expert_weight in place
__global__ void k_softmax_aw(const float* __restrict__ ew, float* __restrict__ ws) {
  __shared__ float red[256];
  const int eb = blockIdx.x, e = eb >> 2, b = eb & 3, tid = threadIdx.x;
  float* L = ws + OFF_AW + (size_t)eb * Tz;
  float l[8], mx = -3.4e38f;
#pragma unroll
  for (int j = 0; j < 8; ++j) { l[j] = L[tid + 256 * j]; mx = fmaxf(mx, l[j]); }
  red[tid] = mx; __syncthreads();
  for (int s = 128; s > 0; s >>= 1) {
    if (tid < s) red[tid] = fmaxf(red[tid], red[tid + s]);
    __syncthreads();
  }
  mx = red[0]; __syncthreads();
  float sum = 0.f;
#pragma unroll
  for (int j = 0; j < 8; ++j) { l[j] = expf(l[j] - mx); sum += l[j]; }
  red[tid] = sum; __syncthreads();
  for (int s = 128; s > 0; s >>= 1) {
    if (tid < s) red[tid] += red[tid + s];
    __syncthreads();
  }
  const float inv = 1.0f / red[0];
#pragma unroll
  for (int j = 0; j < 8; ++j) {
    const int t = tid + 256 * j;
    L[t] = l[j] * inv * ew[(size_t)(b * Tz + t) * Ez + e];
  }
}

// K4: xa_part[eb,chunk,d] = sum_{t in chunk} aw[eb,t] * eo[eb,t,d]
__global__ void k_xa_part(const float* __restrict__ eo, float* __restrict__ ws) {
  const int d = blockIdx.x * 256 + threadIdx.x;
  const int chunk = blockIdx.y, eb = blockIdx.z;
  const float* aw = ws + OFF_AW + (size_t)eb * Tz + chunk * 256;
  const float* x = eo + ((size_t)eb * Tz + chunk * 256) * Dz + d;
  float acc = 0.f;
  for (int t = 0; t < 256; ++t) acc += aw[t] * x[(size_t)t * Dz];
  ws[OFF_XAP + ((size_t)(eb * 8 + chunk) << 10) + d] = acc;
}

__global__ void k_xa_reduce(float* __restrict__ ws) {
  const int gid = blockIdx.x * 256 + threadIdx.x;  // eb*1024 + d
  if (gid >= 24 * Dz) return;
  const int eb = gid >> 10, d = gid & 1023;
  float acc = 0.f;
#pragma unroll
  for (int c = 0; c < 8; ++c) acc += ws[OFF_XAP + ((size_t)(eb * 8 + c) << 10) + d];
  ws[OFF_XA + gid] = acc;
}

// K5: formulas[e,b,f] = xa[e,b].Wv[e][:,f] + bv[e,f]
__global__ void k_formulas(const float* __restrict__ Wv, const float* __restrict__ bv,
                           float* __restrict__ ws) {
  __shared__ float xs[Bz * Dz];
  const int e = blockIdx.y, f = blockIdx.x * 256 + threadIdx.x;
  for (int i = threadIdx.x; i < Bz * Dz; i += 256) xs[i] = ws[OFF_XA + e * Bz * Dz + i];
  __syncthreads();
  float a0 = 0.f, a1 = 0.f, a2 = 0.f, a3 = 0.f;
  const float* W = Wv + ((size_t)e << 20) + f;
  for (int d = 0; d < Dz; ++d) {
    const float w = W[(size_t)d << 10];
    a0 += xs[d] * w; a1 += xs[Dz + d] * w;
    a2 += xs[2 * Dz + d] * w; a3 += xs[3 * Dz + d] * w;
  }
  const float bias = bv[e * Dz + f];
  float* out = ws + OFF_FORM + (size_t)e * Bz * Dz + f;
  out[0] = a0 + bias; out[Dz] = a1 + bias;
  out[2 * Dz] = a2 + bias; out[3 * Dz] = a3 + bias;
}

// K6: c[p] = [fa@Wa+ba | fb@Wb+bb]   (WMMA)
__global__ void k_pair_ab(const float* __restrict__ Wa, const float* __restrict__ ba,
                          const float* __restrict__ Wb, const float* __restrict__ bb,
                          float* __restrict__ ws) {
  __shared__ float xs[Bz * Dz];
  const int tile = blockIdx.x;          // 0..7 ; tiles 0-3 => A-half, 4-7 => B-half
  const int p = blockIdx.y;
  const int half = tile >> 2;
  const int ei = half ? (p % Ez) : (p / Ez);
  for (int i = threadIdx.x; i < Bz * Dz; i += 256)
    xs[i] = ws[OFF_FORM + (size_t)ei * Bz * Dz + i];
  __syncthreads();
  const float* W = (half ? Wb : Wa) + (size_t)p * Dz * DAz;
  const float* bias = (half ? bb : ba) + p * DAz;
  const int n0 = (tile & 3) * 128 + (threadIdx.x >> 5) * 16;
  v8f acc = wmma_rowblock(W, DAz, xs, Dz, n0);
  const int lane = threadIdx.x & 31, nidx = lane & 15;
  if ((lane >> 4) == 0) {
    const int col = half * DAz + n0 + nidx;
    const float bv = bias[n0 + nidx];
#pragma unroll
    for (int r = 0; r < Bz; ++r)
      ws[OFF_CMAT + ((size_t)(p * Bz + r) << 10) + col] = acc[r] + bv;
  }
}

// K7: h[p] = gelu(c[p]@gate_W1[p] + b1)   (WMMA)
__global__ void k_gate1(const float* __restrict__ W1, const float* __restrict__ b1,
                        float* __restrict__ ws) {
  __shared__ float xs[Bz * Dz];
  const int p = blockIdx.y;
  for (int i = threadIdx.x; i < Bz * Dz; i += 256)
    xs[i] = ws[OFF_CMAT + ((size_t)p << 12) + i];
  __syncthreads();
  const float* W = W1 + (size_t)p * Dz * DAz;
  const int n0 = blockIdx.x * 128 + (threadIdx.x >> 5) * 16;
  v8f acc = wmma_rowblock(W, DAz, xs, Dz, n0);
  const int lane = threadIdx.x & 31, nidx = lane & 15;
  if ((lane >> 4) == 0) {
    const float bv = b1[p * DAz + n0 + nidx];
#pragma unroll
    for (int r = 0; r < Bz; ++r)
      ws[OFF_H + ((size_t)(p * Bz + r)) * DAz + n0 + nidx] = gelu_exact(acc[r] + bv);
  }
}

// K8: strength, avg_strength, mask
__global__ void k_strength(const float* __restrict__ W2, const float* __restrict__ b2,
                           float* __restrict__ ws, float* __restrict__ out) {
  __shared__ float sL[Bz];
  const int p = blockIdx.x, b = threadIdx.x >> 5, lane = threadIdx.x & 31;
  const float* h = ws + OFF_H + (size_t)(p * Bz + b) * DAz;
  const float* w = W2 + (size_t)p * DAz;
  float acc = 0.f;
  for (int i = 0; i < DAz / 32; ++i) acc += h[lane + 32 * i] * w[lane + 32 * i];
  acc = wave_sum(acc);
  if (lane == 0) {
    const float s = 1.0f / (1.0f + expf(-(acc + b2[p])));
    ws[OFF_STR + p * Bz + b] = s;
    sL[b] = s;
  }
  __syncthreads();
  if (threadIdx.x == 0) {
    const float avg = 0.25f * (sL[0] + sL[1] + sL[2] + sL[3]);
    out[OUT_AVG + p] = avg;
    ws[OFF_MASK + p] = (avg > THRESH) ? 1.0f : 0.0f;
  }
}

__global__ void k_active(float* __restrict__ ws, float* __restrict__ out) {
  __shared__ float red[64];
  const int tid = threadIdx.x;
  red[tid] = (tid < Pz) ? ws[OFF_MASK + tid] : 0.0f;
  __syncthreads();
  for (int s = 32; s > 0; s >>= 1) {
    if (tid < s) red[tid] += red[tid + s];
    __syncthreads();
  }
  if (tid == 0) { ws[OFF_ACT] = red[0]; out[OUT_ACT] = red[0]; }
}

// K9: hs[p] = gelu(c[p]@syn_W1[p] + sb1)   (WMMA)
__global__ void k_syn1(const float* __restrict__ W1, const float* __restrict__ b1,
                       float* __restrict__ ws) {
  __shared__ float xs[Bz * Dz];
  const int p = blockIdx.y;
  for (int i = threadIdx.x; i < Bz * Dz; i += 256)
    xs[i] = ws[OFF_CMAT + ((size_t)p << 12) + i];
  __syncthreads();
  const float* W = W1 + ((size_t)p << 20);
  const int n0 = blockIdx.x * 128 + (threadIdx.x >> 5) * 16;
  v8f acc = wmma_rowblock(W, Dz, xs, Dz, n0);
  const int lane = threadIdx.x & 31, nidx = lane & 15;
  if ((lane >> 4) == 0) {
    const float bv = b1[p * Dz + n0 + nidx];
#pragma unroll
    for (int r = 0; r < Bz; ++r)
      ws[OFF_HS + ((size_t)(p * Bz + r) << 10) + n0 + nidx] = gelu_exact(acc[r] + bv);
  }
}

// K10: insight_scaled[p,b,d] = mask[p]*gate[p]*strength[p,b]*(hs@syn_W2 + sb2)  (WMMA)
__global__ void k_syn2(const float* __restrict__ W2, const float* __restrict__ b2,
                       const float* __restrict__ pgate, float* __restrict__ ws) {
  __shared__ float xs[Bz * Dz];
  const int p = blockIdx.y;
  for (int i = threadIdx.x; i < Bz * Dz; i += 256)
    xs[i] = ws[OFF_HS + ((size_t)p << 12) + i];
  __syncthreads();
  const float* W = W2 + ((size_t)p << 20);
  const int n0 = blockIdx.x * 128 + (threadIdx.x >> 5) * 16;
  v8f acc = wmma_rowblock(W, Dz, xs, Dz, n0);
  const int lane = threadIdx.x & 31, nidx = lane & 15;
  if ((lane >> 4) == 0) {
    const float fac = ws[OFF_MASK + p] * pgate[p];
    const float bv = b2[p * Dz + n0 + nidx];
#pragma unroll
    for (int r = 0; r < Bz; ++r) {
      const float s = ws[OFF_STR + p * Bz + r];
      ws[OFF_INS + ((size_t)(p * Bz + r) << 10) + n0 + nidx] = (acc[r] + bv) * fac * s;
    }
  }
}

// K11: total = sum_p insight_scaled / max(active,1); LayerNorm -> normed
__global__ void k_total_ln(const float* __restrict__ gamma, const float* __restrict__ beta,
                           float* __restrict__ ws) {
  __shared__ float tot[Dz];
  __shared__ float r1[256], r2[256];
  const int b = blockIdx.x, tid = threadIdx.x;
  const float denom = 1.0f / fmaxf(ws[OFF_ACT], 1.0f);
  float s1 = 0.f, s2 = 0.f;
#pragma unroll
  for (int j = 0; j < 4; ++j) {
    const int d = tid + 256 * j;
    float t = 0.f;
    for (int p = 0; p < Pz; ++p) t += ws[OFF_INS + ((size_t)(p * Bz + b) << 10) + d];
    t *= denom;
    tot[d] = t; s1 += t; s2 += t * t;
  }
  r1[tid] = s1; r2[tid] = s2; __syncthreads();
  for (int s = 128; s > 0; s >>= 1) {
    if (tid < s) { r1[tid] += r1[tid + s]; r2[tid] += r2[tid + s]; }
    __syncthreads();
  }
  const float mean = r1[0] * (1.0f / Dz);
  const float var = r2[0] * (1.0f / Dz) - mean * mean;
  const float inv = rsqrtf(var + LN_EPS);
#pragma unroll
  for (int j = 0; j < 4; ++j) {
    const int d = tid + 256 * j;
    ws[OFF_NRM + b * Dz + d] = (tot[d] - mean) * inv * gamma[d] + beta[d];
  }
}

// K12: projection partials over d-chunks
__global__ void k_proj_part(const float* __restrict__ outW, float* __restrict__ ws) {
  __shared__ float xs[Bz * 256];
  const int f = blockIdx.x * 256 + threadIdx.x, dc = blockIdx.y;
  for (int i = threadIdx.x; i < Bz * 256; i += 256) {
    const int b = i >> 8, j = i & 255;
    xs[i] = ws[OFF_NRM + b * Dz + dc * 256 + j];
  }
  __syncthreads();
  float a0 = 0.f, a1 = 0.f, a2 = 0.f, a3 = 0.f;
  for (int j = 0; j < 256; ++j) {
    const float w = outW[((size_t)(dc * 256 + j) << 10) + f];
    a0 += xs[j] * w; a1 += xs[256 + j] * w;
    a2 += xs[512 + j] * w; a3 += xs[768 + j] * w;
  }
  float* pp = ws + OFF_PPART + ((size_t)dc * Bz) * Dz + f;
  pp[0] = a0; pp[Dz] = a1; pp[2 * Dz] = a2; pp[3 * Dz] = a3;
}

__global__ void k_proj_fin(const float* __restrict__ outb, const float* __restrict__ agate,
                           float* __restrict__ ws) {
  const int f = blockIdx.x * 256 + threadIdx.x;
  const float ag = agate[0], bv = outb[f];
#pragma unroll
  for (int b = 0; b < Bz; ++b) {
    float v = bv;
#pragma unroll
    for (int dc = 0; dc < 4; ++dc) v += ws[OFF_PPART + ((size_t)(dc * Bz + b)) * Dz + f];
    ws[OFF_ADD + b * Dz + f] = v * ag;
  }
}

// K13: output = bridge + (active>0 ? addition : 0), broadcast over T
__global__ void k_final(const float* __restrict__ bridge, const float* __restrict__ ws,
                        float* __restrict__ out) {
  const size_t i4 = (size_t)blockIdx.x * 256 + threadIdx.x;
  const size_t off = i4 * 4;
  const int b = (int)(off >> 21);       // T*D = 2^21
  const int d = (int)(off & (Dz - 1));
  const bool act = ws[OFF_ACT] > 0.0f;
  float4 v = ((const float4*)bridge)[i4];
  if (act) {
    const float4 a = *(const float4*)(ws + OFF_ADD + b * Dz + d);
    v.x += a.x; v.y += a.y; v.z += a.z; v.w += a.w;
  }
  ((float4*)out)[i4] = v;
}

// ===========================================================================
extern "C" void kernel_launch(void* const* d_in, const int* in_sizes, int n_in,
                              void* d_out, int out_size, void* d_ws, size_t ws_size,
                              hipStream_t stream) {
  (void)in_sizes; (void)n_in; (void)out_size; (void)ws_size;
  const float* bridge = (const float*)d_in[0];
  const float* eo     = (const float*)d_in[1];
  const float* ew     = (const float*)d_in[2];
  const float* q      = (const float*)d_in[3];
  const float* Wk     = (const float*)d_in[4];
  const float* bk     = (const float*)d_in[5];
  const float* Wv     = (const float*)d_in[6];
  const float* bv     = (const float*)d_in[7];
  const float* pWa    = (const float*)d_in[8];
  const float* pba    = (const float*)d_in[9];
  const float* pWb    = (const float*)d_in[10];
  const float* pbb    = (const float*)d_in[11];
  const float* gW1    = (const float*)d_in[12];
  const float* gb1    = (const float*)d_in[13];
  const float* gW2    = (const float*)d_in[14];
  const float* gb2    = (const float*)d_in[15];
  const float* sW1    = (const float*)d_in[16];
  const float* sb1    = (const float*)d_in[17];
  const float* sW2    = (const float*)d_in[18];
  const float* sb2    = (const float*)d_in[19];
  const float* pgate  = (const float*)d_in[20];
  const float* gamma  = (const float*)d_in[21];
  const float* beta   = (const float*)d_in[22];
  const float* outW   = (const float*)d_in[23];
  const float* outb   = (const float*)d_in[24];
  const float* agate  = (const float*)d_in[25];
  float* out = (float*)d_out;
  float* ws = (float*)d_ws;

  k_qbias<<<1, 256, 0, stream>>>(q, bk, ws);
  k_fold_q<<<Ez * Dz / 8, 256, 0, stream>>>(Wk, q, ws);
  k_logits<<<Ez * Bz * Tz / 8, 256, 0, stream>>>(eo, ew, ws);
  k_softmax_aw<<<Ez * Bz, 256, 0, stream>>>(ew, ws);
  k_xa_part<<<dim3(Dz / 256, 8, Ez * Bz), 256, 0, stream>>>(eo, ws);
  k_xa_reduce<<<(Ez * Bz * Dz + 255) / 256, 256, 0, stream>>>(ws);
  k_formulas<<<dim3(Dz / 256, Ez), 256, 0, stream>>>(Wv, bv, ws);
  k_pair_ab<<<dim3(8, Pz), 256, 0, stream>>>(pWa, pba, pWb, pbb, ws);
  k_gate1<<<dim3(DAz / 128, Pz), 256, 0, stream>>>(gW1, gb1, ws);
  k_strength<<<Pz, 128, 0, stream>>>(gW2, gb2, ws, out);
  k_active<<<1, 64, 0, stream>>>(ws, out);
  k_syn1<<<dim3(Dz / 128, Pz), 256, 0, stream>>>(sW1, sb1, ws);
  k_syn2<<<dim3(Dz / 128, Pz), 256, 0, stream>>>(sW2, sb2, pgate, ws);
  k_total_ln<<<Bz, 256, 0, stream>>>(gamma, beta, ws);
  k_proj_part<<<dim3(Dz / 256, 4), 256, 0, stream>>>(outW, ws);
  k_proj_fin<<<Dz / 256, 256, 0, stream>>>(outb, agate, ws);
  k_final<<<(Bz * Tz * Dz / 4) / 256, 256, 0, stream>>>(bridge, ws, out);
}